// AdditiveAttention_38869454029308
// MI455X (gfx1250) — compile-verified
//
#include <hip/hip_runtime.h>
#include <math.h>

#define B_ 32
#define T_ 2048
#define D_ 1024

typedef __attribute__((ext_vector_type(2))) float v2f;
typedef __attribute__((ext_vector_type(8))) float v8f;

// Branchless tanh: 1 - 2/(e^{2x}+1).  v_exp_f32 + v_rcp_f32, no EXEC games.
// x->+inf: exp->inf, rcp->0 -> +1.  x->-inf: exp->0, rcp(1)=1 -> -1.
__device__ __forceinline__ float fast_tanh(float x) {
    const float e = __expf(2.0f * x);
    return 1.0f - 2.0f * __builtin_amdgcn_rcpf(e + 1.0f);
}

// ---------------------------------------------------------------------------
// Kernel 1: energies[b,t] = sum_d v[d] * tanh(keys[b,t,d] + query[b,d] + wf[b,t,d])
// One wave computes 16 rows via V_WMMA_F32_16X16X4_F32 accumulation over D.
// K=8 per step via one b128 load per stream; the dot product is invariant under
// K-permutation, so (x,y) feed WMMA#1 and (z,w) feed WMMA#2 with a matching
// broadcast-v B matrix.  A-layout: lanes 0-15 carry K=0,1; lanes 16-31 K=2,3.
// ---------------------------------------------------------------------------
__global__ __launch_bounds__(256) void energies_kernel(
    const float* __restrict__ keys, const float* __restrict__ query,
    const float* __restrict__ wfb,  const float* __restrict__ vv,
    const int* __restrict__ lens,   float* __restrict__ eOut)
{
    const int blk   = blockIdx.x;
    const int b     = blk >> 4;        // 16 chunks of 128 rows per batch
    const int chunk = blk & 15;
    const int wave  = threadIdx.x >> 5;
    const int lane  = threadIdx.x & 31;
    const int t_base = chunk * 128 + wave * 16;
    const int L = lens[b];
    if (t_base >= L) return;           // wave-uniform: masked rows never used

    const int m   = lane & 15;         // row within the 16-row tile
    const int kk  = (lane >> 4) << 1;  // K offset {0,2} for the v2f operands
    const int off = kk << 1;           // float4 byte-group offset {0,4}

    const float* krow = keys + ((size_t)b * T_ + (size_t)(t_base + m)) * D_;
    const float* wrow = wfb  + ((size_t)b * T_ + (size_t)(t_base + m)) * D_;
    const float* q    = query + (size_t)b * D_;

    v8f acc = {};
    #pragma unroll 2
    for (int d0 = 0; d0 < D_; d0 += 8) {
        const int di = d0 + off;
        if ((d0 & 127) == 0) {         // uniform: prefetch streams ahead
            __builtin_prefetch(krow + di + 128, 0, 3);
            __builtin_prefetch(wrow + di + 128, 0, 3);
        }
        const float4 kf = *(const float4*)(krow + di);
        const float4 wf = *(const float4*)(wrow + di);
        const float4 qf = *(const float4*)(q    + di);
        const float4 vf = *(const float4*)(vv   + di);

        v2f a, bb;
        a.x  = fast_tanh(kf.x + qf.x + wf.x);
        a.y  = fast_tanh(kf.y + qf.y + wf.y);
        bb.x = vf.x; bb.y = vf.y;
        acc = __builtin_amdgcn_wmma_f32_16x16x4_f32(
                  false, a, false, bb, (short)0, acc, false, false);

        a.x  = fast_tanh(kf.z + qf.z + wf.z);
        a.y  = fast_tanh(kf.w + qf.w + wf.w);
        bb.x = vf.z; bb.y = vf.w;
        acc = __builtin_amdgcn_wmma_f32_16x16x4_f32(
                  false, a, false, bb, (short)0, acc, false, false);
    }

    // D-matrix layout: lanes 0-15 hold M=0..7 in acc[0..7]; lanes 16-31 hold M=8..15.
    if ((lane & 15) == 0) {
        float* e = eOut + (size_t)b * T_ + t_base + (lane >> 4) * 8;
        e[0] = acc[0]; e[1] = acc[1]; e[2] = acc[2]; e[3] = acc[3];
        e[4] = acc[4]; e[5] = acc[5]; e[6] = acc[6]; e[7] = acc[7];
    }
}

// ---------------------------------------------------------------------------
// Kernel 2: masked softmax over T per batch. One 256-thread block per batch.
// ---------------------------------------------------------------------------
__global__ __launch_bounds__(256) void softmax_kernel(
    const float* __restrict__ eIn, const int* __restrict__ lens,
    float* __restrict__ wOut)
{
    const int b    = blockIdx.x;
    const int tid  = threadIdx.x;
    const int wave = tid >> 5, lane = tid & 31;
    const int L    = lens[b];
    __shared__ float red[8];

    float e[8];
    float mx = -INFINITY;
    #pragma unroll
    for (int i = 0; i < 8; ++i) {
        const int t = i * 256 + tid;
        const float x = (t < L) ? eIn[(size_t)b * T_ + t] : -INFINITY;
        e[i] = x; mx = fmaxf(mx, x);
    }
    #pragma unroll
    for (int off = 16; off > 0; off >>= 1) mx = fmaxf(mx, __shfl_down(mx, off, 32));
    if (lane == 0) red[wave] = mx;
    __syncthreads();
    float bmx = red[0];
    #pragma unroll
    for (int j = 1; j < 8; ++j) bmx = fmaxf(bmx, red[j]);
    __syncthreads();

    float s = 0.f;
    #pragma unroll
    for (int i = 0; i < 8; ++i) { const float p = __expf(e[i] - bmx); e[i] = p; s += p; }
    #pragma unroll
    for (int off = 16; off > 0; off >>= 1) s += __shfl_down(s, off, 32);
    if (lane == 0) red[wave] = s;
    __syncthreads();
    float tot = 0.f;
    #pragma unroll
    for (int j = 0; j < 8; ++j) tot += red[j];
    const float inv = 1.0f / tot;

    #pragma unroll
    for (int i = 0; i < 8; ++i) {
        const int t = i * 256 + tid;
        wOut[(size_t)b * T_ + t] = e[i] * inv;   // 0 for masked t
    }
}

// ---------------------------------------------------------------------------
// Kernel 3: context[b,d] = sum_{t<L} weights[b,t] * value[b,t,d]
// Weights staged in LDS; thread-per-column, fully coalesced value reads.
// ---------------------------------------------------------------------------
__global__ __launch_bounds__(256) void context_kernel(
    const float* __restrict__ value, const float* __restrict__ weights,
    const int* __restrict__ lens,    float* __restrict__ ctx)
{
    const int b     = blockIdx.x >> 2;
    const int chunk = blockIdx.x & 3;
    const int tid   = threadIdx.x;
    const int d     = chunk * 256 + tid;
    const int L     = lens[b];

    __shared__ float sW[T_];
    #pragma unroll
    for (int i = 0; i < 8; ++i)
        sW[i * 256 + tid] = weights[(size_t)b * T_ + i * 256 + tid];
    __syncthreads();

    const float* vcol = value + (size_t)b * T_ * D_ + d;
    float acc = 0.f;
    int t = 0;
    for (; t + 7 < L; t += 8) {
        __builtin_prefetch(vcol + (size_t)(t + 16) * D_, 0, 3);
        float a0 = sW[t]     * vcol[(size_t)t * D_];
        float a1 = sW[t + 1] * vcol[(size_t)(t + 1) * D_];
        float a2 = sW[t + 2] * vcol[(size_t)(t + 2) * D_];
        float a3 = sW[t + 3] * vcol[(size_t)(t + 3) * D_];
        float a4 = sW[t + 4] * vcol[(size_t)(t + 4) * D_];
        float a5 = sW[t + 5] * vcol[(size_t)(t + 5) * D_];
        float a6 = sW[t + 6] * vcol[(size_t)(t + 6) * D_];
        float a7 = sW[t + 7] * vcol[(size_t)(t + 7) * D_];
        acc += ((a0 + a1) + (a2 + a3)) + ((a4 + a5) + (a6 + a7));
    }
    for (; t < L; ++t) acc += sW[t] * vcol[(size_t)t * D_];
    ctx[(size_t)b * D_ + d] = acc;
}

// ---------------------------------------------------------------------------
extern "C" void kernel_launch(void* const* d_in, const int* in_sizes, int n_in,
                              void* d_out, int out_size, void* d_ws, size_t ws_size,
                              hipStream_t stream)
{
    (void)in_sizes; (void)n_in; (void)out_size; (void)ws_size;
    const float* keys  = (const float*)d_in[0];
    const float* value = (const float*)d_in[1];
    const float* query = (const float*)d_in[2];
    const float* wfb   = (const float*)d_in[3];
    const float* v     = (const float*)d_in[4];
    const int*   lens  = (const int*)d_in[5];

    float* out = (float*)d_out;
    float* ctx = out;                       // [B, D]   (first return value)
    float* wts = out + (size_t)B_ * D_;     // [B, T, 1] (second return value)
    float* eWS = (float*)d_ws;              // [B, T] energies scratch (256 KB)

    energies_kernel<<<B_ * (T_ / 128), 256, 0, stream>>>(keys, query, wfb, v, lens, eWS);
    softmax_kernel<<<B_, 256, 0, stream>>>(eWS, lens, wts);
    context_kernel<<<B_ * 4, 256, 0, stream>>>(value, wts, lens, ctx);
}